// GCNLayer_41609643164180
// MI455X (gfx1250) — compile-verified
//
#include <hip/hip_runtime.h>
#include <hip/hip_bf16.h>

// ---------------------------------------------------------------------------
// GCN layer: h = segment_sum(features[edge_src], edge_dst); out = relu(h@W + b)
// N=100000, E=1600000, D=F=128  (fp32)
//
// Phase 1: zero h (d_ws scratch, N*128 floats)
// Phase 2: wave-per-edge gather + L2-resident native f32 atomic scatter-add
// Phase 3: WMMA f32 16x16x4 GEMM + bias + relu  (memory-bound -> keep fp32)
// ---------------------------------------------------------------------------

typedef __attribute__((ext_vector_type(2))) float v2f;
typedef __attribute__((ext_vector_type(8))) float v8f;

// ---- Phase 1: zero the accumulator buffer --------------------------------
__global__ void gcn_zero_kernel(float4* __restrict__ p, size_t n4) {
    size_t i = (size_t)blockIdx.x * blockDim.x + threadIdx.x;
    const size_t stride = (size_t)gridDim.x * blockDim.x;
    for (; i < n4; i += stride) {
        p[i] = make_float4(0.f, 0.f, 0.f, 0.f);
    }
}

// ---- Phase 2: one wave32 per edge; 512B row gather + 4 atomics/lane ------
__global__ void __launch_bounds__(256)
gcn_scatter_kernel(const float* __restrict__ feat,
                   const int* __restrict__ esrc,
                   const int* __restrict__ edst,
                   float* __restrict__ h, int E) {
    const int edge = blockIdx.x * 8 + (threadIdx.x >> 5);  // 8 waves / block
    if (edge >= E) return;
    const int lane = threadIdx.x & 31;

    // Wave-uniform indices: force scalar so the row bases live in SGPRs and
    // the gather becomes a single SADDR-form global_load_b128.
    const int s = __builtin_amdgcn_readfirstlane(esrc[edge]);
    const int d = __builtin_amdgcn_readfirstlane(edst[edge]);

    const float4 v = *(const float4*)(feat + (size_t)s * 128 + lane * 4);
    float* hp = h + (size_t)d * 128 + lane * 4;

    // Native non-returning fp32 atomics -> global_atomic_add_f32 (L2-resident)
    unsafeAtomicAdd(hp + 0, v.x);
    unsafeAtomicAdd(hp + 1, v.y);
    unsafeAtomicAdd(hp + 2, v.z);
    unsafeAtomicAdd(hp + 3, v.w);
}

// ---- Phase 3: WMMA f32 GEMM (16x16x4), bias + relu -----------------------
// One wave computes a 16x128 output stripe: 8 accumulators (8 VGPRs each),
// A-fragment reused across the 8 N-tiles of every K-chunk.
__global__ void __launch_bounds__(256)
gcn_gemm_bias_relu_kernel(const float* __restrict__ h,
                          const float* __restrict__ W,
                          const float* __restrict__ bias,
                          float* __restrict__ out, int N) {
    const int lane    = threadIdx.x & 31;
    const int wave    = threadIdx.x >> 5;
    const int rowBase = (blockIdx.x * 8 + wave) * 16;
    if (rowBase >= N) return;  // uniform per wave: no partial-EXEC WMMA

    const int half = lane >> 4;   // 0: K pair {0,1}; 1: K pair {2,3}
    const int l15  = lane & 15;

    int arow = rowBase + l15;
    if (arow >= N) arow = N - 1;  // clamp address; results predicated on store
    const float* aptr = h + (size_t)arow * 128 + half * 2;

    v8f acc[8];
#pragma unroll
    for (int t = 0; t < 8; ++t)
        acc[t] = v8f{0.f, 0.f, 0.f, 0.f, 0.f, 0.f, 0.f, 0.f};

#pragma unroll 4
    for (int kc = 0; kc < 32; ++kc) {
        const int k0 = kc * 4;

        // A fragment: h[rowBase + l15][k0 + half*2 + {0,1}]  (contiguous b64)
        v2f a;
        a.x = aptr[k0];
        a.y = aptr[k0 + 1];

        const int krow   = k0 + half * 2;
        const float* wp0 = W + (size_t)krow * 128 + l15;
        const float* wp1 = wp0 + 128;

#pragma unroll
        for (int t = 0; t < 8; ++t) {
            // B fragment: W[krow + {0,1}][t*16 + l15]
            v2f bf;
            bf.x = wp0[t * 16];
            bf.y = wp1[t * 16];
            acc[t] = __builtin_amdgcn_wmma_f32_16x16x4_f32(
                /*neg_a=*/false, a, /*neg_b=*/false, bf,
                /*c_mod=*/(short)0, acc[t],
                /*reuse_a=*/false, /*reuse_b=*/false);
        }
    }

    // C/D layout: VGPR r -> row (rowBase + half*8 + r), col = t*16 + l15
#pragma unroll
    for (int t = 0; t < 8; ++t) {
        const int col  = t * 16 + l15;
        const float bv = bias[col];
#pragma unroll
        for (int r = 0; r < 8; ++r) {
            const int row = rowBase + half * 8 + r;
            if (row < N) {
                const float v = acc[t][r] + bv;
                out[(size_t)row * 128 + col] = v > 0.f ? v : 0.f;
            }
        }
    }
}

// ---------------------------------------------------------------------------
extern "C" void kernel_launch(void* const* d_in, const int* in_sizes, int n_in,
                              void* d_out, int out_size, void* d_ws, size_t ws_size,
                              hipStream_t stream) {
    const float* features = (const float*)d_in[0];  // [N,128]
    const float* W        = (const float*)d_in[1];  // [128,128]
    const float* b        = (const float*)d_in[2];  // [128]
    const int*   esrc     = (const int*)d_in[3];    // [E]
    const int*   edst     = (const int*)d_in[4];    // [E]

    const int N = in_sizes[0] / 128;
    const int E = in_sizes[3];

    float* h   = (float*)d_ws;        // N*128 fp32 accumulator (51.2 MB)
    float* out = (float*)d_out;

    // Phase 1: zero h
    const size_t n4 = (size_t)N * 128 / 4;
    gcn_zero_kernel<<<4096, 256, 0, stream>>>((float4*)h, n4);

    // Phase 2: edge scatter (8 edges per 256-thread block, wave32 per edge)
    const int sblocks = (E + 7) / 8;
    gcn_scatter_kernel<<<sblocks, 256, 0, stream>>>(features, esrc, edst, h, E);

    // Phase 3: GEMM + bias + relu (128 rows per block: 8 waves x 16 rows)
    const int gblocks = (N + 127) / 128;
    gcn_gemm_bias_relu_kernel<<<gblocks, 256, 0, stream>>>(h, W, b, out, N);
}